// UniversalDiffMC_56779467653750
// MI455X (gfx1250) — compile-verified
//
#include <hip/hip_runtime.h>
#include <cstdint>

typedef float f4 __attribute__((ext_vector_type(4)));
typedef float f2 __attribute__((ext_vector_type(2)));

constexpr int   DIM = 128;
constexpr int   NC  = 129;                         // cells per axis (dim+1)
constexpr float OFF = 0.3f;
constexpr float INV = 1.0f / 128.0f;               // exact power of two
constexpr long long NCELLS = (long long)NC * NC * NC;   // 2,146,689
constexpr long long O_TRI  = NCELLS * 12;               // 25,760,268 (verts floats)
constexpr long long O_MSK  = O_TRI + NCELLS * 6;        // 38,640,402

constexpr int TK = 64, TJ = 2, TI = 2;             // cells per block
constexpr int SK = TK + 1, SJ = TJ + 1, SI = TI + 1;  // 65,3,3 gp slab
constexpr int SLAB = SI * SJ * SK;                 // 585 dwords
constexpr int NTHREADS = TK * TJ * TI;             // 256 = 8 waves (wave32)

__global__ __launch_bounds__(NTHREADS)
void mc_kernel(const float* __restrict__ grid,
               const float* __restrict__ deform,
               float* __restrict__ out)
{
    __shared__ float sgp[SI][SJ][SK];

    const int k0 = blockIdx.x * TK;
    const int j0 = blockIdx.y * TJ;
    const int i0 = blockIdx.z * TI;

    const int tid = ((int)threadIdx.z * TJ + (int)threadIdx.y) * TK + (int)threadIdx.x;

    // ---- Stage padded-grid slab into LDS with CDNA5 async global->LDS copies.
    // In-range elements use global_load_async_to_lds_b32 (ASYNCcnt path);
    // pad elements (value iso+1 = 1.0) are written with plain DS stores.
    for (int f = tid; f < SLAB; f += NTHREADS) {
        const int ii = f / (SJ * SK);
        const int r  = f % (SJ * SK);
        const int jj = r / SK;
        const int kk = r % SK;
        const int a = i0 + ii, b = j0 + jj, c = k0 + kk;   // padded coords
        const bool inr = (a >= 1) & (a <= DIM) & (b >= 1) & (b <= DIM) &
                         (c >= 1) & (c <= DIM);
        if (inr) {
            // addrspace(3) offset: low 32 bits of the flat LDS address
            const uint32_t loff = (uint32_t)(uintptr_t)(void*)&sgp[ii][jj][kk];
            const uint64_t ga = (uint64_t)(uintptr_t)(grid +
                ((size_t)(a - 1) * DIM + (size_t)(b - 1)) * DIM + (size_t)(c - 1));
            asm volatile("global_load_async_to_lds_b32 %0, %1, off"
                         :: "v"(loff), "v"(ga) : "memory");
        } else {
            sgp[ii][jj][kk] = 1.0f;   // ISOVALUE + 1 padding
        }
    }
    asm volatile("s_wait_asynccnt 0" ::: "memory");
    __syncthreads();

    const int ti = threadIdx.z, tj = threadIdx.y, tk = threadIdx.x;
    const int i = i0 + ti, j = j0 + tj, k = k0 + tk;
    if (i >= NC || j >= NC || k >= NC) return;

    // 8 cube corners from LDS
    const float c000 = sgp[ti][tj][tk];
    const float c001 = sgp[ti][tj][tk + 1];
    const float c010 = sgp[ti][tj + 1][tk];
    const float c011 = sgp[ti][tj + 1][tk + 1];
    const float c100 = sgp[ti + 1][tj][tk];
    const float c101 = sgp[ti + 1][tj][tk + 1];
    const float c110 = sgp[ti + 1][tj + 1][tk];
    const float c111 = sgp[ti + 1][tj + 1][tk + 1];

    const float mn = fminf(fminf(fminf(c000, c001), fminf(c010, c011)),
                           fminf(fminf(c100, c101), fminf(c110, c111)));
    const float mx = fmaxf(fmaxf(fmaxf(c000, c001), fmaxf(c010, c011)),
                           fmaxf(fmaxf(c100, c101), fmaxf(c110, c111)));
    const float mask = (mn <= 0.0f && 0.0f <= mx) ? 1.0f : 0.0f;

    // deform lookup (zero on the padded boundary layer i/j/k == 0)
    float dx = 0.0f, dy = 0.0f, dz = 0.0f;
    if (i >= 1 && j >= 1 && k >= 1) {
        const float* ds = deform +
            3ull * (((size_t)(i - 1) * DIM + (size_t)(j - 1)) * DIM + (size_t)(k - 1));
        dx = ds[0]; dy = ds[1]; dz = ds[2];
    }

    const float cx = (float)i + 0.5f + dx;
    const float cy = (float)j + 0.5f + dy;
    const float cz = (float)k + 0.5f + dz;

    // post-transform: ((center +/- OFF) - 1)/128 - 1, then mask
    const float axm = (((cx - OFF) - 1.0f) * INV - 1.0f) * mask;
    const float axp = (((cx + OFF) - 1.0f) * INV - 1.0f) * mask;
    const float aym = (((cy - OFF) - 1.0f) * INV - 1.0f) * mask;
    const float ayp = (((cy + OFF) - 1.0f) * INV - 1.0f) * mask;
    const float az  = ((cz - 1.0f) * INV - 1.0f) * mask;

    const long long ci = ((long long)i * NC + j) * NC + k;

    // verts: 12 floats [xm,ym,z, xp,ym,z, xp,yp,z, xm,yp,z] -> 3x b128 NT stores
    float* vp = out + 12 * ci;                 // 48*ci bytes: 16B aligned
    f4 va = {axm, aym, az, axp};
    f4 vb = {aym, az, axp, ayp};
    f4 vc = {az, axm, ayp, az};
    __builtin_nontemporal_store(va, (f4*)(vp + 0));
    __builtin_nontemporal_store(vb, (f4*)(vp + 4));
    __builtin_nontemporal_store(vc, (f4*)(vp + 8));

    // tris: [b, b+1, b+2, b, b+2, b+3] * mask; 4*ci+3 < 2^24 so exact in f32
    const float bidx = (float)(4 * ci);
    float* tp = out + O_TRI + 6 * ci;          // 8B aligned
    f2 ta = {bidx * mask, (bidx + 1.0f) * mask};
    f2 tb = {(bidx + 2.0f) * mask, bidx * mask};
    f2 tc = {(bidx + 2.0f) * mask, (bidx + 3.0f) * mask};
    __builtin_nontemporal_store(ta, (f2*)(tp + 0));
    __builtin_nontemporal_store(tb, (f2*)(tp + 2));
    __builtin_nontemporal_store(tc, (f2*)(tp + 4));

    // vmask: 4 copies (region base is only 8B aligned -> two b64 stores)
    float* mp = out + O_MSK + 4 * ci;
    f2 mm = {mask, mask};
    __builtin_nontemporal_store(mm, (f2*)(mp + 0));
    __builtin_nontemporal_store(mm, (f2*)(mp + 2));
}

extern "C" void kernel_launch(void* const* d_in, const int* in_sizes, int n_in,
                              void* d_out, int out_size, void* d_ws, size_t ws_size,
                              hipStream_t stream) {
    const float* grid   = (const float*)d_in[0];   // [128,128,128] f32
    const float* deform = (const float*)d_in[1];   // [128,128,128,3] f32
    float* out = (float*)d_out;                    // verts | tris | vmask (flat f32)

    dim3 block(TK, TJ, TI);                        // (64,2,2) = 256 threads
    dim3 grd((NC + TK - 1) / TK,                   // 3
             (NC + TJ - 1) / TJ,                   // 65
             (NC + TI - 1) / TI);                  // 65
    mc_kernel<<<grd, block, 0, stream>>>(grid, deform, out);
}